// WarpingLayer_87900800680176
// MI455X (gfx1250) — compile-verified
//
#include <hip/hip_runtime.h>
#include <math.h>

#ifndef __has_builtin
#define __has_builtin(x) 0
#endif

namespace {
constexpr int NB = 8;
constexpr int NC = 16;
constexpr int NH = 512;
constexpr int NW = 512;
constexpr int HW = NH * NW;   // 262144 = 2^18
constexpr int BLOCK = 256;    // 8 wave32 waves
}

// gfx1250 async global->LDS path (ASYNCcnt), guarded so the kernel always compiles.
#if __has_builtin(__builtin_amdgcn_global_load_async_to_lds_b32) && \
    __has_builtin(__builtin_amdgcn_s_wait_asynccnt)
#define USE_ASYNC_LDS 1
#else
#define USE_ASYNC_LDS 0
#endif

__global__ __launch_bounds__(BLOCK) void warp_bilinear_kernel(
    const float* __restrict__ img,
    const float* __restrict__ flo,
    float* __restrict__ out) {
  const int t   = threadIdx.x;
  const int p   = blockIdx.x * BLOCK + t;  // 0 .. NB*HW-1 (2^21)
  const int b   = p >> 18;                 // / HW
  const int rem = p & (HW - 1);            // % HW
  const int y   = rem >> 9;                // / NW
  const int x   = rem & (NW - 1);          // % NW

  // Flow planes for this batch; a block covers 256 consecutive pixels of one batch.
  const float* g0 = flo + (size_t)b * 2 * HW + rem;  // flo[b,0,y,x]
  const float* g1 = g0 + HW;                         // flo[b,1,y,x]

#if USE_ASYNC_LDS
  __shared__ float s_flo[2 * BLOCK];
  {
    // Builtin signature (per hipcc diagnostic): (AS1 int*, AS3 int*, imm offset, imm cpol)
    typedef __attribute__((address_space(1))) int GI;
    typedef __attribute__((address_space(3))) int LI;
    __builtin_amdgcn_global_load_async_to_lds_b32((GI*)g0, (LI*)&s_flo[t], 0, 0);
    __builtin_amdgcn_global_load_async_to_lds_b32((GI*)g1, (LI*)&s_flo[BLOCK + t], 0, 0);
  }
#endif

  // Prefetch next block's flow chunk (global_prefetch_b8); read-only, low locality.
  __builtin_prefetch(g0 + BLOCK, 0, 0);

#if USE_ASYNC_LDS
  __builtin_amdgcn_s_wait_asynccnt(0);  // wave-private: each lane reads its own slot
  const float fx = s_flo[t];
  const float fy = s_flo[BLOCK + t];
#else
  const float fx = __builtin_nontemporal_load(g0);
  const float fy = __builtin_nontemporal_load(g1);
#endif

  // Faithful normalize/unnormalize round-trip (align_corners=True), same op order
  // as the reference so floor()/mask decisions agree.
  const float vx = (float)x + fx;
  const float vy = (float)y + fy;
  const float nx = 2.0f * vx / 511.0f - 1.0f;
  const float ny = 2.0f * vy / 511.0f - 1.0f;
  const float ix = (nx + 1.0f) * 511.0f * 0.5f;
  const float iy = (ny + 1.0f) * 511.0f * 0.5f;

  const float x0 = floorf(ix), y0 = floorf(iy);
  const float x1 = x0 + 1.0f,  y1 = y0 + 1.0f;
  const float wx1 = ix - x0,   wy1 = iy - y0;
  const float wx0 = 1.0f - wx1, wy0 = 1.0f - wy1;

  const bool vxa = (x0 >= 0.0f) && (x0 <= 511.0f);
  const bool vxb = (x1 >= 0.0f) && (x1 <= 511.0f);
  const bool vya = (y0 >= 0.0f) && (y0 <= 511.0f);
  const bool vyb = (y1 >= 0.0f) && (y1 <= 511.0f);

  float w00 = (vxa && vya) ? wx0 * wy0 : 0.0f;
  float w10 = (vxb && vya) ? wx1 * wy0 : 0.0f;
  float w01 = (vxa && vyb) ? wx0 * wy1 : 0.0f;
  float w11 = (vxb && vyb) ? wx1 * wy1 : 0.0f;

  // mask = (sum of valid weights >= 1.0); fold it into the weights.
  const float msum = ((w00 + w10) + w01) + w11;
  const float m = (msum >= 1.0f) ? 1.0f : 0.0f;
  w00 *= m; w10 *= m; w01 *= m; w11 *= m;

  // Clamped integer corner offsets, shared by all 16 channels.
  const int xi0 = (int)fminf(fmaxf(x0, 0.0f), 511.0f);
  const int xi1 = (int)fminf(fmaxf(x1, 0.0f), 511.0f);
  const int yi0 = (int)fminf(fmaxf(y0, 0.0f), 511.0f);
  const int yi1 = (int)fminf(fmaxf(y1, 0.0f), 511.0f);
  const int o00 = yi0 * NW + xi0;
  const int o10 = yi0 * NW + xi1;
  const int o01 = yi1 * NW + xi0;
  const int o11 = yi1 * NW + xi1;

  const float* __restrict__ ib = img + (size_t)b * NC * HW;  // wave-uniform base
  float* __restrict__ ob = out + (size_t)b * NC * HW + rem;

  // Fully unrolled: 64 gathers in flight, 4 shared address VGPRs + SGPR bases.
#pragma unroll
  for (int c = 0; c < NC; ++c) {
    const float* __restrict__ pc = ib + (size_t)c * HW;
    const float v = pc[o00] * w00 + pc[o10] * w10 + pc[o01] * w01 + pc[o11] * w11;
    // out is write-once, never re-read: non-temporal to preserve L2 for img.
    __builtin_nontemporal_store(v, ob + (size_t)c * HW);
  }
}

extern "C" void kernel_launch(void* const* d_in, const int* in_sizes, int n_in,
                              void* d_out, int out_size, void* d_ws, size_t ws_size,
                              hipStream_t stream) {
  (void)in_sizes; (void)n_in; (void)out_size; (void)d_ws; (void)ws_size;
  const float* img = (const float*)d_in[0];
  const float* flo = (const float*)d_in[1];
  float* out = (float*)d_out;

  const int nblocks = (NB * HW) / BLOCK;  // 8192, exact
  warp_bilinear_kernel<<<dim3(nblocks), dim3(BLOCK), 0, stream>>>(img, flo, out);
}